// AnomalyGeneration_62096637165973
// MI455X (gfx1250) — compile-verified
//
#include <hip/hip_runtime.h>

typedef __attribute__((ext_vector_type(2))) float v2f;
typedef __attribute__((ext_vector_type(8))) float v8f;

#define KCB     2048      // codebook entries
#define DDIM    128       // embedding dim
#define SKIPN   102       // int(2048 * 0.05)
#define KSTRIDE 2052      // key row stride (pads so row-halves hit disjoint LDS banks)
#define ZST     132       // z-tile row stride (floats), 8B-aligned, bank-spread

// Dynamic LDS layout (bytes)
#define OFF_KEYS 0
#define OFF_ZT   (16 * KSTRIDE * 4)            // 131328
#define OFF_BINS (OFF_ZT + 16 * ZST * 4)       // +8448  = 139776
#define OFF_SEL  (OFF_BINS + 8 * 256 * 4)      // +8192  = 147968
#define OFF_ML   (OFF_SEL + 16 * 4)            // 148032
#define SMEM_TOTAL (OFF_ML + 16 * 4)           // 148096 bytes < 320KB WGP LDS

// Monotone map: larger float -> smaller key (descending order sorts ascending by key)
static __device__ __forceinline__ unsigned keyDesc(float f) {
    unsigned u = __float_as_uint(f);
    unsigned m = (u & 0x80000000u) ? 0xFFFFFFFFu : 0x80000000u;
    return ~(u ^ m);
}

static __device__ __forceinline__ v8f wmma_f32(v2f a, v2f b, v8f c) {
    return __builtin_amdgcn_wmma_f32_16x16x4_f32(false, a, false, b, (short)0, c,
                                                 false, false);
}

// ---------------------------------------------------------------------------
// Prep: codebook norms + avg-pooled masks -> d_ws
// ws floats: [0,2048) cbnF | [2048,4096) cbnC | [4096,36864) Mf | [36864,45056) Mc
// ---------------------------------------------------------------------------
__global__ __launch_bounds__(256)
void prep_kernel(const float* __restrict__ cbF, const float* __restrict__ cbC,
                 const float* __restrict__ M, float* __restrict__ ws)
{
    int i = blockIdx.x * 256 + threadIdx.x;
    float* cbnF = ws;
    float* cbnC = ws + 2048;
    float* Mf   = ws + 4096;
    float* Mc   = ws + 4096 + 32768;
    if (i < 2048) {
        float s = 0.f;
        for (int d = 0; d < DDIM; ++d) { float v = cbF[(size_t)i * DDIM + d]; s += v * v; }
        cbnF[i] = s;
    } else if (i < 4096) {
        int k = i - 2048;
        float s = 0.f;
        for (int d = 0; d < DDIM; ++d) { float v = cbC[(size_t)k * DDIM + d]; s += v * v; }
        cbnC[k] = s;
    } else if (i < 4096 + 32768) {            // fine mask pool: 4x4 blocks of (128,512)
        int j = i - 4096;
        int b = j >> 12, rem = j & 4095;
        int hl = rem >> 7, wl = rem & 127;
        const float* Mb = M + (size_t)b * 128 * 512;
        float s = 0.f;
        for (int dh = 0; dh < 4; ++dh)
            for (int dw = 0; dw < 4; ++dw)
                s += Mb[(hl * 4 + dh) * 512 + wl * 4 + dw];
        Mf[j] = s * (1.0f / 16.0f);
    } else if (i < 4096 + 32768 + 8192) {     // coarse mask pool: 8x8 blocks
        int j = i - 4096 - 32768;
        int b = j >> 10, rem = j & 1023;
        int hl = rem >> 6, wl = rem & 63;
        const float* Mb = M + (size_t)b * 128 * 512;
        float s = 0.f;
        for (int dh = 0; dh < 8; ++dh)
            for (int dw = 0; dw < 8; ++dw)
                s += Mb[(hl * 8 + dh) * 512 + wl * 8 + dw];
        Mc[j] = s * (1.0f / 64.0f);
    }
}

// ---------------------------------------------------------------------------
// Fused: WMMA f32 score GEMM (16 rows x 2048 cols, K=D=128) -> LDS keys,
// per-row radix-select of the pos-th ranked column, blended gather epilogue.
// One workgroup (8 wave32) per 16 query points. dist tile never leaves the WGP.
// GEMM runs 2 N-tiles x 2 split accumulators = 4 independent WMMA chains;
// the tile-pair loop stays rolled (unroll 1) to keep VGPRs < 256 so 4 waves/SIMD
// fit alongside the 148KB LDS footprint (2 workgroups per 320KB WGP).
// ---------------------------------------------------------------------------
__global__ __launch_bounds__(256)
void distant_kernel(const float* __restrict__ z,        // (B,D,H,W)
                    const float* __restrict__ q,        // (B,D,H,W)
                    const float* __restrict__ cb,       // (K,D)
                    const float* __restrict__ cbnorm,   // (K)
                    const float* __restrict__ Ml,       // (B,HW) pooled mask
                    const float* __restrict__ strength, // (B,)
                    const float* __restrict__ randu,    // (B,HW)
                    float* __restrict__ out,            // (B,D,H,W)
                    int hwShift)                        // HW = 1 << hwShift
{
    extern __shared__ char smem[];
    unsigned* keys = (unsigned*)(smem + OFF_KEYS);   // [16][KSTRIDE]
    float*    zT   = (float*)(smem + OFF_ZT);        // [16][ZST]
    int*      bins = (int*)(smem + OFF_BINS);        // [8][256]
    int*      selS = (int*)(smem + OFF_SEL);         // [16]
    float*    mlS  = (float*)(smem + OFF_ML);        // [16]

    const int HW   = 1 << hwShift;
    const int tid  = threadIdx.x;
    const int lane = tid & 31;
    const int wave = tid >> 5;

    const int rowBase = blockIdx.x * 16;        // global query row in [0, B*HW)
    const int b       = rowBase >> hwShift;     // HW % 16 == 0: tile never crosses batch
    const int nBase   = rowBase & (HW - 1);

    // ---- stage z tile (16 rows x 128 dims) into LDS; coalesced over w
    for (int i = tid; i < 16 * DDIM; i += 256) {
        int d = i >> 4, r = i & 15;
        zT[r * ZST + d] = z[((size_t)(b * DDIM + d)) * HW + nBase + r];
    }
    __syncthreads();

    // ---- preload all A fragments for the D-loop (ISA 16x4 f32 A layout:
    // v0: K=0 (lanes 0-15) / K=2 (lanes 16-31); v1: K=1 / K=3)
    const int koff = (lane >> 4) << 1;
    v2f afrag[32];
    {
        int r = lane & 15;
        #pragma unroll
        for (int kk = 0; kk < 32; ++kk)
            afrag[kk] = *(const v2f*)&zT[r * ZST + kk * 4 + koff];
    }

    // ---- GEMM: each wave owns 16 contiguous N-tiles; pairs of tiles per pass
    #pragma unroll 1
    for (int tp = 0; tp < 8; ++tp) {
        const int t0 = wave * 16 + tp * 2;
        const int nA = t0 * 16 + (lane & 15);
        const int nB = nA + 16;
        const float* bpA = cb + (size_t)nA * DDIM + koff;
        const float* bpB = cb + (size_t)nB * DDIM + koff;
        v8f cA0 = {}, cA1 = {}, cB0 = {}, cB1 = {};
        #pragma unroll
        for (int kk = 0; kk < 32; kk += 2) {
            v2f bA0 = *(const v2f*)(bpA + kk * 4);
            v2f bB0 = *(const v2f*)(bpB + kk * 4);
            v2f bA1 = *(const v2f*)(bpA + kk * 4 + 4);
            v2f bB1 = *(const v2f*)(bpB + kk * 4 + 4);
            cA0 = wmma_f32(afrag[kk],     bA0, cA0);
            cB0 = wmma_f32(afrag[kk],     bB0, cB0);
            cA1 = wmma_f32(afrag[kk + 1], bA1, cA1);
            cB1 = wmma_f32(afrag[kk + 1], bB1, cB1);
        }
        v8f cA = cA0 + cA1;
        v8f cB = cB0 + cB1;
        // score = ||cb||^2 - 2 z.cb  (per-row ||z||^2 is rank-invariant, dropped)
        const float cbnA = cbnorm[nA];
        const float cbnB = cbnorm[nB];
        const int   rAdd = (lane >> 4) << 3;    // C layout: vgpr r -> M=r / M=r+8
        #pragma unroll
        for (int rr = 0; rr < 8; ++rr) {
            keys[(rr + rAdd) * KSTRIDE + nA] = keyDesc(cbnA - 2.0f * cA[rr]);
            keys[(rr + rAdd) * KSTRIDE + nB] = keyDesc(cbnB - 2.0f * cB[rr]);
        }
    }
    __syncthreads();

    // ---- per-row exact radix select: key48 = keyDesc(score) << 11 | col
    // (stable tie-break by column, matching argsort). One wave per row, 2 rows/wave.
    int* myBins = bins + wave * 256;
    for (int rpass = 0; rpass < 2; ++rpass) {
        const int r    = wave * 2 + rpass;
        const int gRow = nBase + r;

        float st  = strength[b];
        int   T   = (int)floorf(st * (float)KCB);
        T = min(max(T, 1), KCB - 1);
        int span  = max(T - SKIPN, 1);
        float ru  = randu[(size_t)b * HW + gRow];
        int pos   = SKIPN + (int)floorf(ru * (float)span);
        pos = min(max(pos, 0), KCB - 1);

        // cache this wave's 64 strided keys in registers (A-frag regs are dead now)
        const unsigned* krow = &keys[r * KSTRIDE];
        unsigned kreg[64];
        #pragma unroll
        for (int j = 0; j < 64; ++j) kreg[j] = krow[lane + j * 32];

        unsigned long long prefix = 0ull;
        for (int pass = 0; pass < 6; ++pass) {
            const int shift = 40 - 8 * pass;
            for (int j2 = lane; j2 < 256; j2 += 32) myBins[j2] = 0;
            __syncthreads();
            #pragma unroll
            for (int j = 0; j < 64; ++j) {
                unsigned long long key =
                    ((unsigned long long)kreg[j] << 11) | (unsigned)(lane + j * 32);
                if ((key >> (shift + 8)) == (prefix >> (shift + 8)))
                    atomicAdd(&myBins[(int)((key >> shift) & 0xFF)], 1);
            }
            __syncthreads();
            // find the bin containing rank 'pos': 8 bins/lane + wave scan
            int cnt[8]; int lsum = 0;
            #pragma unroll
            for (int j2 = 0; j2 < 8; ++j2) { cnt[j2] = myBins[lane * 8 + j2]; lsum += cnt[j2]; }
            int incl = lsum;
            #pragma unroll
            for (int off = 1; off < 32; off <<= 1) {
                int v = __shfl_up(incl, off, 32);
                if (lane >= off) incl += v;
            }
            int excl = incl - lsum;
            int found = -1, pIn = 0;
            if (pos >= excl && pos < excl + lsum) {
                int p = pos - excl;
                #pragma unroll
                for (int j2 = 0; j2 < 8; ++j2) {
                    if (found < 0) {
                        if (p < cnt[j2]) { found = lane * 8 + j2; pIn = p; }
                        else p -= cnt[j2];
                    }
                }
            }
            unsigned long long mb = __ballot(found >= 0);
            int src  = __ffsll((unsigned long long)mb) - 1;
            int bsel = __shfl(found, src, 32);
            pos      = __shfl(pIn, src, 32);
            prefix  |= ((unsigned long long)(unsigned)bsel) << shift;
            __syncthreads();
        }
        if (lane == 0) {
            selS[r] = (int)(prefix & 0x7FFull);          // column index recovered from key
            mlS[r]  = Ml[(size_t)b * HW + gRow];
        }
    }
    __syncthreads();

    // ---- epilogue: out[b,:,h,w] = ml*cb[sel] + (1-ml)*q ; 16 rows x 16 d-lanes
    {
        const int r  = tid & 15;
        const int dl = tid >> 4;
        const int gRow = nBase + r;
        const int sel  = selS[r];
        const float ml = mlS[r];
        const size_t base = (size_t)b * DDIM * HW + gRow;
        for (int d = dl; d < DDIM; d += 16) {
            float e  = cb[(size_t)sel * DDIM + d];
            float qv = q[base + (size_t)d * HW];
            out[base + (size_t)d * HW] = ml * e + (1.0f - ml) * qv;
        }
    }
}

// ---------------------------------------------------------------------------
extern "C" void kernel_launch(void* const* d_in, const int* in_sizes, int n_in,
                              void* d_out, int out_size, void* d_ws, size_t ws_size,
                              hipStream_t stream) {
    const float* q_fine   = (const float*)d_in[0];   // (8,128,32,128)
    const float* q_coarse = (const float*)d_in[1];   // (8,128,16,64)
    const float* M        = (const float*)d_in[2];   // (8,1,128,512)
    const float* cb_fine  = (const float*)d_in[3];   // (2048,128)
    const float* cb_coarse= (const float*)d_in[4];
    const float* z_fine   = (const float*)d_in[5];
    const float* z_coarse = (const float*)d_in[6];
    const float* s_fine   = (const float*)d_in[7];   // (8,)
    const float* s_coarse = (const float*)d_in[8];
    const float* r_fine   = (const float*)d_in[9];   // (8,4096)
    const float* r_coarse = (const float*)d_in[10];  // (8,1024)

    float* ws   = (float*)d_ws;
    float* cbnF = ws;
    float* cbnC = ws + 2048;
    float* Mf   = ws + 4096;
    float* Mc   = ws + 4096 + 32768;

    float* outF = (float*)d_out;                     // fine output first
    float* outC = outF + (size_t)8 * 128 * 32 * 128; // then coarse

    hipFuncSetAttribute((const void*)distant_kernel,
                        hipFuncAttributeMaxDynamicSharedMemorySize, SMEM_TOTAL);

    // prep: 2048 + 2048 + 32768 + 8192 = 45056 jobs
    prep_kernel<<<(45056 + 255) / 256, 256, 0, stream>>>(cb_fine, cb_coarse, M, ws);

    const int HWF = 32 * 128, HWC = 16 * 64;         // 4096 (shift 12), 1024 (shift 10)
    distant_kernel<<<(8 * HWF) / 16, 256, SMEM_TOTAL, stream>>>(
        z_fine, q_fine, cb_fine, cbnF, Mf, s_fine, r_fine, outF, 12);
    distant_kernel<<<(8 * HWC) / 16, 256, SMEM_TOTAL, stream>>>(
        z_coarse, q_coarse, cb_coarse, cbnC, Mc, s_coarse, r_coarse, outC, 10);
}